// NerfModel_84061099917408
// MI455X (gfx1250) — compile-verified
//
#include <hip/hip_runtime.h>

// NeRF voxel-grid + SH eval for MI455X (gfx1250, wave32).
//
// Roofline: 192^3 x 28 f32 grid = 792 MB >> 192 MB L2, random point gathers
// => ~0.6-0.9 GB of gather traffic, ~40-60 us floor at 23.3 TB/s. ~250 FMA
// per point => 5 orders of magnitude under the WMMA ceiling, and the per-point
// contractions use *private* matrices (no shared operand), so WMMA/TDM-tile
// GEMM paths do not apply. Optimization targets:
//   - 16B-aligned global_load_b128 gathers (28 floats == 7 x float4 per cell),
//     batched per (dz,dy) row-pair so the scheduler issues 14-load clauses.
//   - CDNA5 async global->LDS double buffering (ASYNCcnt) for the streaming
//     x/d inputs: each wave stages its next 32-point chunk while computing
//     the current one; s_wait_asynccnt <= 2 retires the older chunk (async
//     loads complete in order) with no barriers (wave-private staging).
//     Staged data is consumed via addrspace(3) pointers => ds_load, keeping
//     the DS pipe separate from the LOADcnt gather traffic.
//   - Non-temporal stores for the write-once outputs.

#define GLOBAL_AS __attribute__((address_space(1)))
#define LDS_AS    __attribute__((address_space(3)))

typedef int v4i __attribute__((ext_vector_type(4)));

namespace {
constexpr float kScale = 1.5f;
constexpr int   kN     = 192;
constexpr int   kC     = 28;
constexpr int   kBlock = 256;

constexpr float SH_C0   = 0.28209479177387814f;
constexpr float SH_C1   = 0.4886025119029199f;
constexpr float SH_C2_0 = 1.0925484305920792f;
constexpr float SH_C2_1 = -1.0925484305920792f;
constexpr float SH_C2_2 = 0.31539156525252005f;
constexpr float SH_C2_3 = -1.0925484305920792f;
constexpr float SH_C2_4 = 0.5462742152960396f;

__device__ __forceinline__ int clampi(int v, int lo, int hi) {
  return v < lo ? lo : (v > hi ? hi : v);
}
} // namespace

__global__ __launch_bounds__(kBlock)
void nerf_voxel_sh_kernel(const float* __restrict__ xs,
                          const float* __restrict__ dirs,
                          const float* __restrict__ grid,
                          float* __restrict__ out_color,
                          float* __restrict__ out_sigma,
                          int P) {
  // Double-buffered staging: per buffer, 256 points x 3 floats per array.
  __shared__ float sx[2][kBlock * 3];
  __shared__ float sd[2][kBlock * 3];

  const int tid    = (int)threadIdx.x;
  const int lane   = tid & 31;
  const int wave   = tid >> 5;
  const int stride = (int)(gridDim.x * blockDim.x);
  const int i0     = (int)blockIdx.x * kBlock + tid;
  const int wf0    = (int)blockIdx.x * kBlock + wave * 32; // wave's first point

  // Stage one 32-point chunk (x and d) for this wave into buffer `buf`.
  // 32 points * 3 floats = 96 floats = 24 float4 per array; lanes 0..23 each
  // move one 16B-aligned float4 via the async global->LDS path (ASYNCcnt).
  auto stage = [&](int wfp, int buf) {
    if (lane < 24) {
      const float* gx = xs   + 3 * wfp;   // 12*wfp bytes, wfp%32==0 => 16B aligned
      const float* gd = dirs + 3 * wfp;
      float* lx = &sx[buf][wave * 96];
      float* ld = &sd[buf][wave * 96];
      __builtin_amdgcn_global_load_async_to_lds_b128(
          (GLOBAL_AS v4i*)(gx + 4 * lane), (LDS_AS v4i*)(lx + 4 * lane), 0, 0);
      __builtin_amdgcn_global_load_async_to_lds_b128(
          (GLOBAL_AS v4i*)(gd + 4 * lane), (LDS_AS v4i*)(ld + 4 * lane), 0, 0);
    }
  };

  // Prologue: stage chunk 0 if the wave's whole chunk is in range.
  const bool full0 = (wf0 + 32 <= P);
  if (full0) stage(wf0, 0);

  int wfp = wf0;
  int k   = 0;
  for (int i = i0; i < P; i += stride, ++k, wfp += stride) {
    const int  buf        = k & 1;
    const bool full_cur   = (wfp + 32 <= P);
    const int  wfp_next   = wfp + stride;
    const bool stage_next = (wfp_next + 32 <= P);

    // Kick off next chunk's async copy before consuming the current one.
    if (stage_next) stage(wfp_next, buf ^ 1);

    float x0, x1, x2, dvx, dvy, dvz;
    if (full_cur) {
      // Retire the current chunk: with the next chunk's 2 ops possibly in
      // flight, asynccnt <= 2 guarantees the older (in-order) pair landed.
      if (stage_next) __builtin_amdgcn_s_wait_asynccnt(2);
      else            __builtin_amdgcn_s_wait_asynccnt(0);
      // Compiler-level fence: LDS reads below must not hoist above the wait
      // (the async writes are invisible to alias analysis).
      asm volatile("" ::: "memory");
      // addrspace(3)-qualified loads => ds_load_b32 (DS pipe, DScnt), not
      // flat/global loads.
      LDS_AS const float* lx =
          (LDS_AS const float*)&sx[buf][wave * 96 + lane * 3];
      LDS_AS const float* ld =
          (LDS_AS const float*)&sd[buf][wave * 96 + lane * 3];
      x0 = lx[0]; x1 = lx[1]; x2 = lx[2];
      dvx = ld[0]; dvy = ld[1]; dvz = ld[2];
    } else {
      // Partial tail chunk: direct global loads.
      x0 = xs[3 * i + 0]; x1 = xs[3 * i + 1]; x2 = xs[3 * i + 2];
      dvx = dirs[3 * i + 0]; dvy = dirs[3 * i + 1]; dvz = dirs[3 * i + 2];
    }

    // Strict cube mask, same comparison set as the reference.
    const bool inside = (x0 < kScale) && (x1 > -kScale) && (x2 > -kScale) &&
                        (x0 > -kScale) && (x1 < kScale) && (x2 < kScale);

    // indices = arange(P)*mask => masked-out i>0 slots stay zero; slot 0
    // always receives f(x[0], d[0]) (all duplicate writes carry that value).
    if (!inside && i != 0) {
      __builtin_nontemporal_store(0.0f, out_color + 3 * i + 0);
      __builtin_nontemporal_store(0.0f, out_color + 3 * i + 1);
      __builtin_nontemporal_store(0.0f, out_color + 3 * i + 2);
      __builtin_nontemporal_store(0.0f, out_sigma + i);
      continue;
    }

    // grid_sample coords: p = ((x/scale + 1)*N - 1)/2 ; x->W, y->H, z->D.
    const float px = ((x0 / kScale + 1.0f) * (float)kN - 1.0f) * 0.5f;
    const float py = ((x1 / kScale + 1.0f) * (float)kN - 1.0f) * 0.5f;
    const float pz = ((x2 / kScale + 1.0f) * (float)kN - 1.0f) * 0.5f;

    const float fxf = floorf(px), fyf = floorf(py), fzf = floorf(pz);
    const float fx = px - fxf, fy = py - fyf, fz = pz - fzf;
    const int ix = (int)fxf, iy = (int)fyf, iz = (int)fzf;

    const int  cx0 = clampi(ix, 0, kN - 1);
    const int  cx1 = clampi(ix + 1, 0, kN - 1);
    const bool vx0 = (ix >= 0) && (ix < kN);
    const bool vx1 = (ix + 1 >= 0) && (ix + 1 < kN);
    const float wx0 = 1.0f - fx, wx1 = fx;

    float4 acc[7];
#pragma unroll
    for (int c = 0; c < 7; ++c) acc[c] = make_float4(0.0f, 0.0f, 0.0f, 0.0f);

#pragma unroll
    for (int dz = 0; dz < 2; ++dz) {
      const int  izc = iz + dz;
      const bool vz  = (izc >= 0) && (izc < kN);
      const int  cz  = clampi(izc, 0, kN - 1);
      const float wz = dz ? fz : 1.0f - fz;
#pragma unroll
      for (int dy = 0; dy < 2; ++dy) {
        const int  iyc = iy + dy;
        const bool vy  = (iyc >= 0) && (iyc < kN);
        const int  cy  = clampi(iyc, 0, kN - 1);
        const float wy = dy ? fy : 1.0f - fy;

        const int rowbase = (cz * kN + cy) * kN;
        // 28 floats per cell = 112 B = 7 x 16B-aligned float4 => b128 loads.
        const float4* g0 = (const float4*)(grid + (rowbase + cx0) * kC);
        const float4* g1 = (const float4*)(grid + (rowbase + cx1) * kC);

        const float w0 = ((wx0 * wy) * wz) * ((vx0 && vy && vz) ? 1.0f : 0.0f);
        const float w1 = ((wx1 * wy) * wz) * ((vx1 && vy && vz) ? 1.0f : 0.0f);

        // Batch all 14 b128 loads of this row-pair before any FMA so the
        // scheduler can emit one load clause + a single wait.
        float4 a[7], b[7];
#pragma unroll
        for (int c = 0; c < 7; ++c) { a[c] = g0[c]; b[c] = g1[c]; }
#pragma unroll
        for (int c = 0; c < 7; ++c) {
          acc[c].x = fmaf(w1, b[c].x, fmaf(w0, a[c].x, acc[c].x));
          acc[c].y = fmaf(w1, b[c].y, fmaf(w0, a[c].y, acc[c].y));
          acc[c].z = fmaf(w1, b[c].z, fmaf(w0, a[c].z, acc[c].z));
          acc[c].w = fmaf(w1, b[c].w, fmaf(w0, a[c].w, acc[c].w));
        }
      }
    }

    // Spill the 28 interpolated channels to named registers.
    float t[28];
#pragma unroll
    for (int c = 0; c < 7; ++c) {
      t[4 * c + 0] = acc[c].x;
      t[4 * c + 1] = acc[c].y;
      t[4 * c + 2] = acc[c].z;
      t[4 * c + 3] = acc[c].w;
    }

    // Degree-2 SH basis from the view direction.
    const float sxx = dvx * dvx, syy = dvy * dvy, szz = dvz * dvz;
    const float sxy = dvx * dvy, syz = dvy * dvz, sxz = dvx * dvz;

#pragma unroll
    for (int r = 0; r < 3; ++r) {
      const int kb = 1 + 9 * r;
      float res = SH_C0 * t[kb + 0]
                - SH_C1 * dvy * t[kb + 1]
                + SH_C1 * dvz * t[kb + 2]
                - SH_C1 * dvx * t[kb + 3]
                + SH_C2_0 * sxy * t[kb + 4]
                + SH_C2_1 * syz * t[kb + 5]
                + SH_C2_2 * (2.0f * szz - sxx - syy) * t[kb + 6]
                + SH_C2_3 * sxz * t[kb + 7]
                + SH_C2_4 * (sxx - syy) * t[kb + 8];
      __builtin_nontemporal_store(res, out_color + 3 * i + r);
    }
    __builtin_nontemporal_store(fmaxf(t[0], 0.0f), out_sigma + i);
  }
}

extern "C" void kernel_launch(void* const* d_in, const int* in_sizes, int n_in,
                              void* d_out, int out_size, void* d_ws, size_t ws_size,
                              hipStream_t stream) {
  (void)n_in; (void)out_size; (void)d_ws; (void)ws_size;
  const float* xs   = (const float*)d_in[0];   // [P,3] f32
  const float* dirs = (const float*)d_in[1];   // [P,3] f32
  const float* grid = (const float*)d_in[2];   // [192,192,192,28] f32

  const int P = in_sizes[0] / 3;
  float* out_color = (float*)d_out;             // [P,3] f32
  float* out_sigma = out_color + (size_t)3 * P; // [P] f32

  // 1024 blocks x 256 threads (8 wave32/block): 262144 threads, ~4 points
  // each via grid-stride => thousands of waves hide gather latency, and the
  // async LDS staging of chunk k+1 overlaps the gathers of chunk k.
  nerf_voxel_sh_kernel<<<1024, kBlock, 0, stream>>>(xs, dirs, grid,
                                                    out_color, out_sigma, P);
}